// Vim2_74345883893855
// MI455X (gfx1250) — compile-verified
//
#include <hip/hip_runtime.h>
#include <hip/hip_bf16.h>
#include <math.h>

#define DIM 2048
#define LAYER 11

typedef float v2f __attribute__((ext_vector_type(2)));
typedef float v8f __attribute__((ext_vector_type(8)));

// ---------------- LayerNorm: 3 rows of DIM ----------------
__global__ __launch_bounds__(256) void ln_kernel(const float* __restrict__ x,
                                                 const float* __restrict__ g,
                                                 const float* __restrict__ b,
                                                 float* __restrict__ xn) {
    const int row = blockIdx.x;            // 0..2
    const float* xr = x + row * DIM;
    __shared__ float red[256];

    float s = 0.f;
    for (int i = threadIdx.x; i < DIM; i += 256) s += xr[i];
    red[threadIdx.x] = s;
    __syncthreads();
    for (int off = 128; off > 0; off >>= 1) {
        if (threadIdx.x < off) red[threadIdx.x] += red[threadIdx.x + off];
        __syncthreads();
    }
    const float mu = red[0] * (1.0f / DIM);
    __syncthreads();

    float v = 0.f;
    for (int i = threadIdx.x; i < DIM; i += 256) {
        float d = xr[i] - mu;
        v += d * d;
    }
    red[threadIdx.x] = v;
    __syncthreads();
    for (int off = 128; off > 0; off >>= 1) {
        if (threadIdx.x < off) red[threadIdx.x] += red[threadIdx.x + off];
        __syncthreads();
    }
    const float rstd = rsqrtf(red[0] * (1.0f / DIM) + 1e-5f);

    for (int i = threadIdx.x; i < DIM; i += 256)
        xn[row * DIM + i] = (xr[i] - mu) * rstd * g[i] + b[i];
}

// ---------------- WMMA GEMV: out[3][N] = in[3][DIM] * W[N][DIM]^T + bias ----------------
// A = in padded to 16 rows (rows 3..15 zero), B = W tile, fp32 16x16x4 WMMA.
__global__ __launch_bounds__(256) void wmma_gemv_kernel(const float* __restrict__ in,
                                                        const float* __restrict__ W,
                                                        const float* __restrict__ bias,
                                                        float* __restrict__ out,
                                                        int N) {
    // 4 rows in LDS: rows 0..2 = activations, row 3 = zeros (for padded M lanes)
    __shared__ __align__(16) float sIn[4 * DIM];
    for (int i = threadIdx.x; i < 4 * DIM; i += 256)
        sIn[i] = (i < 3 * DIM) ? in[i] : 0.0f;
    __syncthreads();

    const int lane = threadIdx.x & 31;
    const int wave = threadIdx.x >> 5;               // 0..7
    const int tile = blockIdx.x * 8 + wave;          // 16-column output tile
    if (tile * 16 >= N) return;
    const int e0 = tile * 16;

    const int nn    = lane & 15;                     // N index (B frag) / M index (A frag)
    const int khalf = (lane >> 4) * 2;               // K sub-offset per ISA layout
    const int m_eff = (nn < 3) ? nn : 3;             // padded rows read the zero row

    // per-lane base pointers (8-byte aligned: khalf in {0,2}, k step 4)
    const float2* wrow = reinterpret_cast<const float2*>(W + (size_t)(e0 + nn) * DIM + khalf);
    const float2* arow = reinterpret_cast<const float2*>(sIn + m_eff * DIM + khalf);

    v8f acc = {0.f, 0.f, 0.f, 0.f, 0.f, 0.f, 0.f, 0.f};

#pragma unroll 4
    for (int k = 0; k < DIM; k += 4) {
        float2 av = arow[k >> 1];     // A[m][k+khalf], A[m][k+khalf+1]
        float2 wv = wrow[k >> 1];     // W[e0+nn][k+khalf], +1
        v2f a; a.x = av.x; a.y = av.y;
        v2f bfr; bfr.x = wv.x; bfr.y = wv.y;
        acc = __builtin_amdgcn_wmma_f32_16x16x4_f32(
            /*neg_a=*/false, a, /*neg_b=*/false, bfr,
            /*c_mod=*/(short)0, acc, /*reuse_a=*/false, /*reuse_b=*/false);
    }

    // C/D layout: VGPR g, lanes 0-15 => M=g, N=lane. Rows 0..2 in acc[0..2].
    if (lane < 16) {
        const float bv = bias ? bias[e0 + nn] : 0.0f;
        out[0 * N + e0 + nn] = acc[0] + bv;
        out[1 * N + e0 + nn] = acc[1] + bv;
        out[2 * N + e0 + nn] = acc[2] + bv;
    }
}

// ---------------- Fused SSM (seq=1) + silu + residual ----------------
__device__ __forceinline__ float softplus_f(float v) {
    return fmaxf(v, 0.f) + log1pf(__expf(-fabsf(v)));
}

__global__ __launch_bounds__(256) void ssm_final_kernel(const float* __restrict__ z1,
                                                        const float* __restrict__ f,
                                                        const float* __restrict__ bw,
                                                        const float* __restrict__ dbc_f,
                                                        const float* __restrict__ dbc_b,
                                                        const float* __restrict__ dtp_w,
                                                        const float* __restrict__ dtp_b,
                                                        const float* __restrict__ Dp,
                                                        const float* __restrict__ x,
                                                        float* __restrict__ out) {
    __shared__ float sdf[3 * 64];
    __shared__ float sdb[3 * 64];
    for (int i = threadIdx.x; i < 192; i += 256) {
        sdf[i] = dbc_f[i];
        sdb[i] = dbc_b[i];
    }
    __syncthreads();

    const int d = blockIdx.x * 256 + threadIdx.x;
    if (d >= DIM) return;

    float wr[32];
#pragma unroll
    for (int r = 0; r < 32; r++) wr[r] = dtp_w[d * 32 + r];
    const float tb = dtp_b[d];
    const float dp = Dp[d];

#pragma unroll
    for (int b = 0; b < 3; b++) {
        const float* df = sdf + b * 64;
        const float* db = sdb + b * 64;
        // bc = dot(B, C), B = dbc[32:48], C = dbc[48:64]
        float bcf = 0.f, bcb = 0.f;
#pragma unroll
        for (int n = 0; n < 16; n++) {
            bcf += df[32 + n] * df[48 + n];
            bcb += db[32 + n] * db[48 + n];
        }
        float accf = tb, accb = tb;
#pragma unroll
        for (int r = 0; r < 32; r++) {
            accf += df[r] * wr[r];
            accb += db[r] * wr[r];
        }
        const float deltaf = softplus_f(accf);
        const float deltab = softplus_f(accb);
        const float uf = f[b * DIM + d];
        const float ub = bw[b * DIM + d];
        // seq=1, h0=0 => y = u * (delta * dot(B,C) + Dp)
        const float y = uf * (deltaf * bcf + dp) + ub * (deltab * bcb + dp);
        const float zv = z1[b * DIM + d];
        const float z = zv / (1.0f + __expf(-zv));     // silu
        out[b * DIM + d] = y * z + x[b * DIM + d];     // + skip
    }
}

extern "C" void kernel_launch(void* const* d_in, const int* in_sizes, int n_in,
                              void* d_out, int out_size, void* d_ws, size_t ws_size,
                              hipStream_t stream) {
    const float* x       = (const float*)d_in[0];
    const float* norm_g  = (const float*)d_in[1]  + (size_t)LAYER * DIM;
    const float* norm_b  = (const float*)d_in[2]  + (size_t)LAYER * DIM;
    const float* proj_w  = (const float*)d_in[3]  + (size_t)LAYER * DIM * DIM;
    const float* proj_b  = (const float*)d_in[4]  + (size_t)LAYER * DIM;
    const float* fconv_w = (const float*)d_in[5]  + (size_t)LAYER * DIM * DIM;
    const float* fconv_b = (const float*)d_in[6]  + (size_t)LAYER * DIM;
    const float* bconv_w = (const float*)d_in[7]  + (size_t)LAYER * DIM * DIM;
    const float* bconv_b = (const float*)d_in[8]  + (size_t)LAYER * DIM;
    const float* dbc_w   = (const float*)d_in[9]  + (size_t)LAYER * 64 * DIM;
    const float* dtp_w   = (const float*)d_in[10] + (size_t)LAYER * DIM * 32;
    const float* dtp_b   = (const float*)d_in[11] + (size_t)LAYER * DIM;
    // d_in[12] = A_log: mathematically unused (seq len 1, h0 = 0)
    const float* Dp      = (const float*)d_in[13] + (size_t)LAYER * DIM;

    float* ws   = (float*)d_ws;
    float* xn   = ws;                 // 3*DIM
    float* z1   = ws + 3 * DIM;       // 3*DIM
    float* f    = ws + 6 * DIM;       // 3*DIM
    float* bwv  = ws + 9 * DIM;       // 3*DIM
    float* dbcf = ws + 12 * DIM;      // 3*64
    float* dbcb = dbcf + 192;         // 3*64

    ln_kernel<<<3, 256, 0, stream>>>(x, norm_g, norm_b, xn);

    // 2048 cols / (8 waves * 16 cols) = 16 blocks
    wmma_gemv_kernel<<<16, 256, 0, stream>>>(xn, proj_w, proj_b, z1, DIM);
    wmma_gemv_kernel<<<16, 256, 0, stream>>>(z1, fconv_w, fconv_b, f, DIM);
    wmma_gemv_kernel<<<16, 256, 0, stream>>>(z1, bconv_w, bconv_b, bwv, DIM);

    // dbc = u @ dbc_w^T : N = 64 (4 tiles, 1 block)
    wmma_gemv_kernel<<<1, 256, 0, stream>>>(f,   dbc_w, nullptr, dbcf, 64);
    wmma_gemv_kernel<<<1, 256, 0, stream>>>(bwv, dbc_w, nullptr, dbcb, 64);

    ssm_final_kernel<<<DIM / 256, 256, 0, stream>>>(z1, f, bwv, dbcf, dbcb,
                                                    dtp_w, dtp_b, Dp, x, (float*)d_out);
}